// SAGE_71365176590743
// MI455X (gfx1250) — compile-verified
//
#include <hip/hip_runtime.h>
#include <stdint.h>

#define N_NODES   50000
#define N_EDGES   1600000
#define FEATS     128
#define NCLASSES  47
#define SW_LD     132          // LDS row stride (floats): bank-conflict-free b64 reads

typedef __attribute__((ext_vector_type(2))) float v2f;
typedef __attribute__((ext_vector_type(8))) float v8f;

// ---------------- degree: one thread per edge ----------------
__global__ void deg_kernel(const int* __restrict__ dst, float* __restrict__ deg) {
    int e = blockIdx.x * blockDim.x + threadIdx.x;
    if (e < N_EDGES) atomicAdd(&deg[dst[e]], 1.0f);
}

// ---------------- scatter-add: one wave32 per edge ----------------
__global__ void scatter_kernel(const float* __restrict__ feat,
                               const int* __restrict__ src,
                               const int* __restrict__ dst,
                               float* __restrict__ neigh) {
    int wave = (blockIdx.x * blockDim.x + threadIdx.x) >> 5;
    int lane = threadIdx.x & 31;
    if (wave >= N_EDGES) return;
    int s = src[wave];
    int d = dst[wave];
    const float* fs = feat + (size_t)s * FEATS;
    float*       nd = neigh + (size_t)d * FEATS;
#pragma unroll
    for (int j = 0; j < 4; ++j)
        atomicAdd(&nd[lane + 32 * j], fs[lane + 32 * j]);
}

// ---------------- normalize in place: h = (neigh + feat) / (deg + 1) ----------------
__global__ void norm_kernel(float* __restrict__ neigh,
                            const float* __restrict__ feat,
                            const float* __restrict__ deg) {
    int i = blockIdx.x * blockDim.x + threadIdx.x;   // over N_NODES * 128 / 4
    if (i >= N_NODES * (FEATS / 4)) return;
    int node = i / (FEATS / 4);
    float inv = 1.0f / (deg[node] + 1.0f);
    float4* np = ((float4*)neigh) + i;
    const float4* fp = ((const float4*)feat) + i;
    float4 a = *np;
    float4 b = *fp;
    a.x = (a.x + b.x) * inv;
    a.y = (a.y + b.y) * inv;
    a.z = (a.z + b.z) * inv;
    a.w = (a.w + b.w) * inv;
    *np = a;
}

// ---------------- GEMM: out[N_NODES, NOUT] = A[N_NODES,128] @ W[NOUT,128]^T + bias ----------------
// Block = 8 waves. W staged in LDS once per block; each wave owns one 16-row
// tile and register-blocks all column tiles. NOUT is compile-time so all
// clamps/bounds fold away and LDS addresses are base + immediate offset.
template <int NOUT, bool RELU>
__global__ void gemm_kernel(const float* __restrict__ A,
                            const float* __restrict__ W,
                            const float* __restrict__ bias,
                            float* __restrict__ out) {
    constexpr int NC = (NOUT + 15) / 16;
    __shared__ float sw[FEATS * SW_LD];

    // cooperative load of W[NOUT,128] into LDS (float4 -> ds_store_b128)
    for (int i = threadIdx.x; i < NOUT * (FEATS / 4); i += blockDim.x) {
        int r  = i / (FEATS / 4);
        int cc = (i % (FEATS / 4)) * 4;
        float4 w4 = *(const float4*)(W + (size_t)r * FEATS + cc);
        *(float4*)(&sw[r * SW_LD + cc]) = w4;
    }
    __syncthreads();

    int rt = blockIdx.x * 8 + (threadIdx.x >> 5);   // row tile for this wave
    if (rt >= N_NODES / 16) return;                 // wave-uniform exit (after barrier)

    int lane = threadIdx.x & 31;
    int half = lane >> 4;                           // 0: K,K+1   1: K+2,K+3
    int lm   = lane & 15;

    const float* Arow = A + (size_t)(rt * 16 + lm) * FEATS + 2 * half;

    // hoisted per-column-tile LDS base pointers (clamped only where needed)
    const float* bp[NC];
    v8f c[NC];
#pragma unroll
    for (int ct = 0; ct < NC; ++ct) {
        int n = ct * 16 + lm;
        bool full = ((ct + 1) * 16 <= NOUT);        // compile-time per ct
        int nc = full ? n : (n < NOUT ? n : NOUT - 1);
        bp[ct] = &sw[nc * SW_LD + 2 * half];
        float bv = (full || n < NOUT) ? bias[nc] : 0.0f;
#pragma unroll
        for (int r = 0; r < 8; ++r) c[ct][r] = bv;
    }

#pragma unroll
    for (int k = 0; k < FEATS; k += 4) {
        v2f a = *(const v2f*)(Arow + k);
#pragma unroll
        for (int ct = 0; ct < NC; ++ct) {
            v2f b = *(const v2f*)(bp[ct] + k);
            c[ct] = __builtin_amdgcn_wmma_f32_16x16x4_f32(
                        /*neg_a=*/false, a, /*neg_b=*/false, b,
                        /*c_mod=*/(short)0, c[ct], /*reuse_a=*/false, /*reuse_b=*/false);
        }
    }

    int rowBase = rt * 16 + half * 8;
#pragma unroll
    for (int ct = 0; ct < NC; ++ct) {
        int n = ct * 16 + lm;
        if ((ct + 1) * 16 <= NOUT || n < NOUT) {
#pragma unroll
            for (int r = 0; r < 8; ++r) {
                float v = c[ct][r];
                if (RELU) v = fmaxf(v, 0.0f);
                out[(size_t)(rowBase + r) * NOUT + n] = v;
            }
        }
    }
}

extern "C" void kernel_launch(void* const* d_in, const int* in_sizes, int n_in,
                              void* d_out, int out_size, void* d_ws, size_t ws_size,
                              hipStream_t stream) {
    const float* inputs = (const float*)d_in[0];
    const int*   src    = (const int*)d_in[1];
    const int*   dst    = (const int*)d_in[2];
    const float* W0     = (const float*)d_in[3];
    const float* b0     = (const float*)d_in[4];
    const float* W1     = (const float*)d_in[5];
    const float* b1     = (const float*)d_in[6];
    const float* W2     = (const float*)d_in[7];
    const float* b2     = (const float*)d_in[8];
    float* out = (float*)d_out;

    float* deg  = (float*)d_ws;                       // 50000 f32
    float* buf1 = deg + N_NODES;                      // 50000*128 f32
    float* buf2 = buf1 + (size_t)N_NODES * FEATS;     // 50000*128 f32

    const size_t featBytes = (size_t)N_NODES * FEATS * sizeof(float);
    const int EDGE_BLK = (N_EDGES + 255) / 256;
    const int SCAT_BLK = (N_EDGES * 32 + 255) / 256;
    const int NORM_BLK = (N_NODES * (FEATS / 4) + 255) / 256;
    const int GEMM_BLK = (N_NODES / 16 + 7) / 8;      // 3125 row tiles / 8 waves = 391

    // degree
    hipMemsetAsync(deg, 0, N_NODES * sizeof(float), stream);
    deg_kernel<<<EDGE_BLK, 256, 0, stream>>>(dst, deg);

    // ---- layer 0: inputs -> buf2 (ReLU) ----
    hipMemsetAsync(buf1, 0, featBytes, stream);
    scatter_kernel<<<SCAT_BLK, 256, 0, stream>>>(inputs, src, dst, buf1);
    norm_kernel<<<NORM_BLK, 256, 0, stream>>>(buf1, inputs, deg);
    gemm_kernel<FEATS, true><<<GEMM_BLK, 256, 0, stream>>>(buf1, W0, b0, buf2);

    // ---- layer 1: buf2 -> buf2 (ReLU), via buf1 aggregate ----
    hipMemsetAsync(buf1, 0, featBytes, stream);
    scatter_kernel<<<SCAT_BLK, 256, 0, stream>>>(buf2, src, dst, buf1);
    norm_kernel<<<NORM_BLK, 256, 0, stream>>>(buf1, buf2, deg);
    gemm_kernel<FEATS, true><<<GEMM_BLK, 256, 0, stream>>>(buf1, W1, b1, buf2);

    // ---- layer 2: buf2 -> d_out (no activation, 47 classes) ----
    hipMemsetAsync(buf1, 0, featBytes, stream);
    scatter_kernel<<<SCAT_BLK, 256, 0, stream>>>(buf2, src, dst, buf1);
    norm_kernel<<<NORM_BLK, 256, 0, stream>>>(buf1, buf2, deg);
    gemm_kernel<NCLASSES, false><<<GEMM_BLK, 256, 0, stream>>>(buf1, W2, b2, out);
}